// FocalLoss_26766236189461
// MI455X (gfx1250) — compile-verified
//
#include <hip/hip_runtime.h>
#include <stdint.h>

// ---- problem constants (from reference: B=4, C=16, D=64, H=128, W=128) ----
#define DHW            (64 * 128 * 128)      // 1,048,576 spatial positions per batch
#define NBATCH         4
#define CCH            16
#define NROWS          (NBATCH * DHW)        // 4,194,304
#define ROWS_PER_BLOCK 1024
#define THREADS        256
#define ROWS_PER_THREAD (ROWS_PER_BLOCK / THREADS)   // 4
#define NBLOCKS        (NROWS / ROWS_PER_BLOCK)      // 4096
#define TILE_BYTES     (CCH * ROWS_PER_BLOCK * 4)    // 64 KiB LDS tile

typedef __attribute__((ext_vector_type(4))) unsigned int u32x4;
typedef __attribute__((ext_vector_type(8))) int          i32x8;
typedef __attribute__((ext_vector_type(4))) int          i32x4;

// ---------------------------------------------------------------------------
// Kernel 1: per-block TDM load of a 16 x 1024 fp32 tile (row stride DHW elems)
// into LDS, per-row focal-loss terms, block reduction to two partial sums.
// Target (int64) loads are issued BEFORE the tensor-wait barrier so they
// overlap the TDM DMA.
// ---------------------------------------------------------------------------
extern "C" __global__ __launch_bounds__(THREADS)
void focal_main(const float* __restrict__ logits,
                const long long* __restrict__ target,
                float* __restrict__ partials)
{
    extern __shared__ float lds[];   // 16*1024 floats; dynamic LDS base -> offset 0

    const unsigned tid      = threadIdx.x;
    const unsigned bid      = blockIdx.x;
    const unsigned tileRow0 = bid * ROWS_PER_BLOCK;      // first global row of tile
    const unsigned b        = tileRow0 / DHW;            // batch index (tile never crosses)
    const unsigned sp0      = tileRow0 % DHW;            // spatial offset within batch

    // ---- overlap: every wave issues its target loads first ----
    int ti[ROWS_PER_THREAD];
#pragma unroll
    for (int j = 0; j < ROWS_PER_THREAD; ++j)
        ti[j] = (int)target[tileRow0 + tid + (unsigned)j * THREADS];

    // ---- Tensor Data Mover: one DMA per block, issued by wave 0 only ----
    if ((tid >> 5) == 0) {
        const uint64_t ga = (uint64_t)(uintptr_t)logits +
                            4ull * ((uint64_t)b * (uint64_t)(CCH * DHW) + (uint64_t)sp0);

        u32x4 g0;
        g0[0] = 1u;                                          // count=1 (valid), user mode
        g0[1] = 0u;                                          // lds_addr = 0 (dynamic LDS base)
        g0[2] = (unsigned)(ga & 0xFFFFFFFFull);              // global_addr[31:0]
        g0[3] = (unsigned)((ga >> 32) & 0x01FFFFFFull)       // global_addr[56:32]
              | (2u << 30);                                  // type = 2 ("image")

        i32x8 g1;
        g1[0] = (2 << 16);                                   // data_size = 4 bytes
        g1[1] = (int)((unsigned)ROWS_PER_BLOCK << 16);       // tensor_dim0[15:0] = 1024
        g1[2] = (int)((unsigned)CCH << 16);                  // tensor_dim0 hi=0 | tensor_dim1 lo = 16
        g1[3] = (int)((unsigned)ROWS_PER_BLOCK << 16);       // tensor_dim1 hi=0 | tile_dim0 = 1024
        g1[4] = CCH;                                         // tile_dim1 = 16, tile_dim2 = 0
        g1[5] = DHW;                                         // tensor_dim0_stride[31:0]
        g1[6] = 0;                                           // stride hi, tensor_dim1_stride lo
        g1[7] = 0;

        i32x4 g2;
        g2[0] = 1;                                           // tensor_dim2 = 1 (benign)
        g2[1] = 1;                                           // tensor_dim3 = 1
        g2[2] = 0;
        g2[3] = 0;                                           // tile_dim3 = 0 (unused)

        i32x4 g3;
        g3[0] = 0;                                           // tensor_dim3_stride
        g3[1] = (1 << 16);                                   // tensor_dim4 = 1
        g3[2] = 0;                                           // tile_dim4 = 0
        g3[3] = 0;

        i32x8 g4;                                            // extra group (clang-23 6-arg form)
        g4[0] = 0; g4[1] = 0; g4[2] = 0; g4[3] = 0;
        g4[4] = 0; g4[5] = 0; g4[6] = 0; g4[7] = 0;

        __builtin_amdgcn_tensor_load_to_lds(g0, g1, g2, g3, g4, 0);
        __builtin_amdgcn_s_wait_tensorcnt(0);
    }
    __syncthreads();

    // ---- per-row focal terms from LDS ----
    float accL = 0.0f;   // sum fw * ce
    float accW = 0.0f;   // sum fw
#pragma unroll
    for (int j = 0; j < ROWS_PER_THREAD; ++j) {
        const unsigned idx = tid + (unsigned)j * THREADS;    // row within tile
        const int t = ti[j];

        float v[CCH];
#pragma unroll
        for (int c = 0; c < CCH; ++c) v[c] = lds[(unsigned)c * ROWS_PER_BLOCK + idx];

        float m = v[0];
#pragma unroll
        for (int c = 1; c < CCH; ++c) m = fmaxf(m, v[c]);

        float s = 0.0f;
#pragma unroll
        for (int c = 0; c < CCH; ++c) s += __expf(v[c] - m);

        const float xt = lds[(unsigned)t * ROWS_PER_BLOCK + idx];
        const float ce = (m - xt) + __logf(s);               // -log_softmax at target
        const float om = 1.0f - xt;                          // raw-logit "prob" per reference
        const float fw = om * om;                            // gamma = 2
        accL = fmaf(fw, ce, accL);
        accW += fw;
    }

    // ---- block reduction (reuse first 512 floats of the tile after barrier) ----
    __syncthreads();
    lds[tid]           = accL;
    lds[THREADS + tid] = accW;
    __syncthreads();
#pragma unroll
    for (int s = THREADS / 2; s > 0; s >>= 1) {
        if (tid < (unsigned)s) {
            lds[tid]           += lds[tid + s];
            lds[THREADS + tid] += lds[THREADS + tid + s];
        }
        __syncthreads();
    }
    if (tid == 0) {
        partials[2u * bid]     = lds[0];
        partials[2u * bid + 1] = lds[THREADS];
    }
}

// ---------------------------------------------------------------------------
// Kernel 2: reduce 4096 partial pairs, write scalar ratio.
// ---------------------------------------------------------------------------
extern "C" __global__ __launch_bounds__(256)
void focal_final(const float* __restrict__ partials, float* __restrict__ out)
{
    __shared__ float rL[256];
    __shared__ float rW[256];
    float a = 0.0f, w = 0.0f;
    for (unsigned i = threadIdx.x; i < NBLOCKS; i += 256) {
        a += partials[2u * i];
        w += partials[2u * i + 1];
    }
    rL[threadIdx.x] = a;
    rW[threadIdx.x] = w;
    __syncthreads();
#pragma unroll
    for (int s = 128; s > 0; s >>= 1) {
        if (threadIdx.x < (unsigned)s) {
            rL[threadIdx.x] += rL[threadIdx.x + s];
            rW[threadIdx.x] += rW[threadIdx.x + s];
        }
        __syncthreads();
    }
    if (threadIdx.x == 0) out[0] = rL[0] / rW[0];
}

// ---------------------------------------------------------------------------
extern "C" void kernel_launch(void* const* d_in, const int* in_sizes, int n_in,
                              void* d_out, int out_size, void* d_ws, size_t ws_size,
                              hipStream_t stream)
{
    const float*     logits = (const float*)d_in[0];      // [4,16,64,128,128] fp32
    const long long* tgt    = (const long long*)d_in[1];  // [4,64,128,128] int64
    float* partials = (float*)d_ws;                       // 2 * 4096 floats = 32 KiB

    focal_main<<<NBLOCKS, THREADS, TILE_BYTES, stream>>>(logits, tgt, partials);
    focal_final<<<1, 256, 0, stream>>>(partials, (float*)d_out);
}